// DecoderLayer_55327768707350
// MI455X (gfx1250) — compile-verified
//
#include <hip/hip_runtime.h>
#include <math.h>

// ---------------------------------------------------------------------------
// MI455X (gfx1250) decoder layer, fp32 end-to-end using V_WMMA_F32_16X16X4_F32.
// Roofline: ~120 GFLOP; weights (48MB) + activations largely fit in 192MB L2,
// HBM floor is one compulsory pass (~250MB -> ~11us at 23.3 TB/s). GEMMs are
// LDS-staged + double-buffered so global loads are coalesced 128b and overlap
// the WMMA chains instead of load->wait0->mma serialization.
// ---------------------------------------------------------------------------

#define T_SEQ  4096
#define DMODEL 1024
#define NH     16
#define HD     64
#define DMLP   4096
#define WIN    512
#define EPS_F  1.1920929e-07f

// GEMM tiling
#define BM   32
#define BN   256
#define KC   32
#define LSTR 36  // padded LDS row stride (floats); 36*lm mod 64 all-distinct

typedef float v2f __attribute__((ext_vector_type(2)));
typedef float v8f __attribute__((ext_vector_type(8)));

// D = A(16x4) * B(4x16) + C, fp32. (neg_a,A,neg_b,B,c_mod,C,reuse_a,reuse_b)
__device__ __forceinline__ v8f wmma4(v2f a, v2f b, v8f c) {
  return __builtin_amdgcn_wmma_f32_16x16x4_f32(false, a, false, b, (short)0, c,
                                               false, false);
}

// ---------------------------------------------------------------------------
// RMSNorm over rows of length ncol (fp32, eps matches torch F.rms_norm)
// ---------------------------------------------------------------------------
__global__ __launch_bounds__(256) void rmsnorm_kernel(
    const float* __restrict__ in, float* __restrict__ out, int ncol) {
  const int row = blockIdx.x;
  const float* r = in + (size_t)row * ncol;
  float* o = out + (size_t)row * ncol;
  float ss = 0.0f;
  for (int i = threadIdx.x; i < ncol; i += blockDim.x) {
    float v = r[i];
    ss += v * v;
  }
  __shared__ float red[256];
  red[threadIdx.x] = ss;
  __syncthreads();
  for (int s = 128; s > 0; s >>= 1) {
    if (threadIdx.x < (unsigned)s) red[threadIdx.x] += red[threadIdx.x + s];
    __syncthreads();
  }
  float inv = rsqrtf(red[0] / (float)ncol + EPS_F);
  for (int i = threadIdx.x; i < ncol; i += blockDim.x) o[i] = r[i] * inv;
}

// ---------------------------------------------------------------------------
// GEMM: out[m,n] = act( (sum_k A[m,k]*Wt[n,k]) * colScale[n] ) + resid[m,n]
// Wt row-major [N,K] (A @ Wt^T). act==1 -> squared ReLU.
// Block: 256 threads = 8 waves; 32x256 output tile; K-chunks of 32 staged in
// double-buffered LDS with coalesced float4 global loads. Each wave computes
// a 16x64 sub-tile via 4 independent 16x16 fp32 WMMA accumulators.
// Fragment layouts per CDNA5 ISA 7.12.2:
//   A frag : M = lane&15,         K = kk + 2*(lane>>4) + r
//   B frag : N = lane&15,         K = kk + 2*(lane>>4) + r
//   C frag : M = r + 8*(lane>>4), N = lane&15
// ---------------------------------------------------------------------------
__global__ __launch_bounds__(256) void gemm_wt_kernel(
    const float* __restrict__ A, const float* __restrict__ Wt,
    const float* __restrict__ colScale, const float* __restrict__ resid,
    float* __restrict__ out, int M, int N, int K, int act) {
  __shared__ float lA[2][BM * LSTR];
  __shared__ float lW[2][BN * LSTR];
  (void)M;

  const int tid = threadIdx.x;
  const int lane = tid & 31;
  const int wave = tid >> 5;
  const int lm = lane & 15;
  const int half = lane >> 4;
  const int wm = (wave & 1) * 16;   // wave's 16-row half of the 32-row tile
  const int wn = (wave >> 1) * 64;  // wave's 64-col group of the 256-col tile

  const int m0 = blockIdx.y * BM;
  const int n00 = blockIdx.x * BN;

  // cooperative staging map: row = tid>>3 (+32*i), col = (tid&7)*4
  const int ldRow = tid >> 3;
  const int ldCol = (tid & 7) * 4;
  const float* Ag = A + (size_t)(m0 + ldRow) * K + ldCol;
  const float* Wg = Wt + (size_t)(n00 + ldRow) * K + ldCol;

  float4 ra;
  float4 rw[8];

  // prologue: stage chunk 0
  ra = *(const float4*)(Ag);
#pragma unroll
  for (int i = 0; i < 8; ++i)
    rw[i] = *(const float4*)(Wg + (size_t)(i * 32) * K);
  *(float4*)&lA[0][ldRow * LSTR + ldCol] = ra;
#pragma unroll
  for (int i = 0; i < 8; ++i)
    *(float4*)&lW[0][(ldRow + i * 32) * LSTR + ldCol] = rw[i];
  __syncthreads();

  v8f acc[4] = {};
  int cur = 0;
  for (int kc = 0; kc < K; kc += KC) {
    const int nxt = kc + KC;
    if (nxt < K) {  // issue next chunk's global loads before computing
      ra = *(const float4*)(Ag + nxt);
#pragma unroll
      for (int i = 0; i < 8; ++i)
        rw[i] = *(const float4*)(Wg + (size_t)(i * 32) * K + nxt);
    }
    if (kc + 2 * KC < K)  // stream hint two chunks ahead
      __builtin_prefetch(Wg + (size_t)(kc + 2 * KC), 0, 1);

    const float* As = &lA[cur][(wm + lm) * LSTR + 2 * half];
    const float* Ws = &lW[cur][(wn + lm) * LSTR + 2 * half];
#pragma unroll
    for (int kk = 0; kk < KC; kk += 4) {
      v2f a = *(const v2f*)(As + kk);
      v2f b0 = *(const v2f*)(Ws + kk);
      v2f b1 = *(const v2f*)(Ws + kk + 16 * LSTR);
      v2f b2 = *(const v2f*)(Ws + kk + 32 * LSTR);
      v2f b3 = *(const v2f*)(Ws + kk + 48 * LSTR);
      acc[0] = wmma4(a, b0, acc[0]);
      acc[1] = wmma4(a, b1, acc[1]);
      acc[2] = wmma4(a, b2, acc[2]);
      acc[3] = wmma4(a, b3, acc[3]);
    }
    if (nxt < K) {  // store into the other buffer (safe: last read before the previous iter's barrier)
      *(float4*)&lA[cur ^ 1][ldRow * LSTR + ldCol] = ra;
#pragma unroll
      for (int i = 0; i < 8; ++i)
        *(float4*)&lW[cur ^ 1][(ldRow + i * 32) * LSTR + ldCol] = rw[i];
    }
    __syncthreads();
    cur ^= 1;
  }

  const int mB = m0 + wm;
  const int nB = n00 + wn;
#pragma unroll
  for (int s = 0; s < 4; ++s) {
    const int n = nB + s * 16 + lm;
    const float cs = colScale ? colScale[n] : 1.0f;
#pragma unroll
    for (int r = 0; r < 8; ++r) {
      const int m = mB + r + 8 * half;
      float v = acc[s][r] * cs;
      if (act == 1) {
        v = fmaxf(v, 0.0f);
        v = v * v;
      }
      if (resid) v += resid[(size_t)m * N + n];
      out[(size_t)m * N + n] = v;
    }
  }
}

// ---------------------------------------------------------------------------
// Per-head RMSNorm of q,k + partial rotary; repack q,k,v head-major
// qkv: [T, 3*DMODEL] viewed as [T, 3, NH, HD]; outputs [NH, T, HD]
// ---------------------------------------------------------------------------
__global__ __launch_bounds__(64) void qk_rope_kernel(
    const float* __restrict__ qkv, float* __restrict__ qh,
    float* __restrict__ kh, float* __restrict__ vh) {
  const int t = blockIdx.x;
  const int h = blockIdx.y;
  const int d = threadIdx.x;  // 0..63
  const float* base = qkv + (size_t)t * (3 * DMODEL) + h * HD;
  float qv = base[d];
  float kv = base[DMODEL + d];
  float vv = base[2 * DMODEL + d];

  __shared__ float red[64], sq[64], sk[64];
  red[d] = qv * qv;
  __syncthreads();
  for (int s = 32; s > 0; s >>= 1) {
    if (d < s) red[d] += red[d + s];
    __syncthreads();
  }
  float qi = rsqrtf(red[0] / (float)HD + EPS_F);
  __syncthreads();
  red[d] = kv * kv;
  __syncthreads();
  for (int s = 32; s > 0; s >>= 1) {
    if (d < s) red[d] += red[d + s];
    __syncthreads();
  }
  float ki = rsqrtf(red[0] / (float)HD + EPS_F);
  __syncthreads();

  sq[d] = qv * qi;
  sk[d] = kv * ki;
  __syncthreads();

  // partial rotary: freqs over HD/4=16 entries (exponent i/15), rest identity
  const int j = d & 31;
  float theta = (j < 16) ? (float)t * powf(10000.0f, -(float)j / 15.0f) : 0.0f;
  float c = cosf(theta), s_ = sinf(theta);
  float x1q = sq[j], x2q = sq[j + 32];
  float x1k = sk[j], x2k = sk[j + 32];
  float oq = (d < 32) ? (x1q * c + x2q * s_) : (-x1q * s_ + x2q * c);
  float ok = (d < 32) ? (x1k * c + x2k * s_) : (-x1k * s_ + x2k * c);

  const size_t oidx = ((size_t)h * T_SEQ + t) * HD + d;
  qh[oidx] = oq;
  kh[oidx] = ok;
  vh[oidx] = vv;
}

// ---------------------------------------------------------------------------
// Sliding-window flash attention. One wave per (head, 16-query tile).
// Mask (from reference): qp - WIN < kp <= qp, kp >= 0.
// All K/V fragments for a key tile are preloaded into registers before the
// WMMA chains (waits drain incrementally); scores use two accumulators to
// halve the D->C dependency chain on the matrix pipe.
// ---------------------------------------------------------------------------
__global__ __launch_bounds__(32) void attn_kernel(
    const float* __restrict__ Qh, const float* __restrict__ Kh,
    const float* __restrict__ Vh, float* __restrict__ outAttn) {
  const int h = blockIdx.x;
  const int q0 = blockIdx.y * 16;
  const int lane = threadIdx.x;
  const int lm = lane & 15;
  const int half = lane >> 4;
  const size_t hb = (size_t)h * T_SEQ * HD;

  // preload whole Q tile as 16 A-fragments (d-chunks of 4)
  const float* qrow = Qh + hb + (size_t)(q0 + lm) * HD + 2 * half;
  v2f qa[16];
#pragma unroll
  for (int dc = 0; dc < 16; ++dc) qa[dc] = *(const v2f*)(qrow + 4 * dc);

  float mi[8], li[8];
  v8f o[4] = {};
#pragma unroll
  for (int r = 0; r < 8; ++r) {
    mi[r] = -INFINITY;
    li[r] = 0.0f;
  }

  __shared__ float sP[16 * 16];

  const int lo = q0 - WIN + 1;
  const int kt0 = (lo < 0) ? 0 : (lo / 16) * 16;
  for (int kt = kt0; kt <= q0; kt += 16) {
    // ---- S = Q * K^T (16x16): preload 16 B-frags, then chain on 2 accums
    const float* krow = Kh + hb + (size_t)(kt + lm) * HD + 2 * half;
    v2f kb[16];
#pragma unroll
    for (int dc = 0; dc < 16; ++dc) kb[dc] = *(const v2f*)(krow + 4 * dc);
    v8f s0 = {}, s1 = {};
#pragma unroll
    for (int dc = 0; dc < 16; dc += 2) {
      s0 = wmma4(qa[dc], kb[dc], s0);
      s1 = wmma4(qa[dc + 1], kb[dc + 1], s1);
    }
    v8f s = s0 + s1;

    // ---- scale + mask + online softmax (rows live on 16-lane halves)
#pragma unroll
    for (int r = 0; r < 8; ++r) {
      const int qp = q0 + r + 8 * half;
      const int kp = kt + lm;
      float sv = s[r] * 0.125f;  // 1/sqrt(64)
      const bool okm = (kp <= qp) && (kp > qp - WIN);
      sv = okm ? sv : -INFINITY;
      float rm = sv;
      rm = fmaxf(rm, __shfl_xor(rm, 1, 32));
      rm = fmaxf(rm, __shfl_xor(rm, 2, 32));
      rm = fmaxf(rm, __shfl_xor(rm, 4, 32));
      rm = fmaxf(rm, __shfl_xor(rm, 8, 32));
      const float nm = fmaxf(mi[r], rm);
      const float p = (sv == -INFINITY) ? 0.0f : __expf(sv - nm);
      float corr;
      if (nm == -INFINITY)
        corr = 1.0f;  // fully-masked so far: keep O=0, l=0
      else if (mi[r] == -INFINITY)
        corr = 0.0f;
      else
        corr = __expf(mi[r] - nm);
      float rs = p;
      rs += __shfl_xor(rs, 1, 32);
      rs += __shfl_xor(rs, 2, 32);
      rs += __shfl_xor(rs, 4, 32);
      rs += __shfl_xor(rs, 8, 32);
      li[r] = li[r] * corr + rs;
      mi[r] = nm;
      o[0][r] *= corr;
      o[1][r] *= corr;
      o[2][r] *= corr;
      o[3][r] *= corr;
      sP[(r + 8 * half) * 16 + lm] = p;  // stage P (C-layout -> LDS tile)
    }
    __syncthreads();

    // ---- O += P * V : preload all 16 V B-frags + 4 P A-frags, then chain
    const float* vbase = Vh + hb;
    v2f vb[16];
#pragma unroll
    for (int kc4 = 0; kc4 < 4; ++kc4) {
      const int kr = kt + kc4 * 4 + 2 * half;
#pragma unroll
      for (int f = 0; f < 4; ++f) {
        v2f b;
        b.x = vbase[(size_t)kr * HD + f * 16 + lm];
        b.y = vbase[(size_t)(kr + 1) * HD + f * 16 + lm];
        vb[kc4 * 4 + f] = b;
      }
    }
    v2f pa[4];
#pragma unroll
    for (int kc4 = 0; kc4 < 4; ++kc4) {
      pa[kc4].x = sP[lm * 16 + kc4 * 4 + 2 * half];
      pa[kc4].y = sP[lm * 16 + kc4 * 4 + 2 * half + 1];
    }
#pragma unroll
    for (int kc4 = 0; kc4 < 4; ++kc4)
#pragma unroll
      for (int f = 0; f < 4; ++f) o[f] = wmma4(pa[kc4], vb[kc4 * 4 + f], o[f]);
    __syncthreads();
  }

  // normalize rows and scatter back to [T, DMODEL] (col = h*64 + d)
#pragma unroll
  for (int f = 0; f < 4; ++f) {
#pragma unroll
    for (int r = 0; r < 8; ++r) {
      const int tq = q0 + r + 8 * half;
      const int col = h * HD + f * 16 + lm;
      outAttn[(size_t)tq * DMODEL + col] = o[f][r] / li[r];
    }
  }
}

// ---------------------------------------------------------------------------
// Launch: rmsnorm -> QKV gemm -> qk-norm/rope repack -> attention ->
//         o-proj(+scale,+resid) -> rmsnorm -> w1(relu^2) -> w2(+scale,+resid)
// ---------------------------------------------------------------------------
extern "C" void kernel_launch(void* const* d_in, const int* in_sizes, int n_in,
                              void* d_out, int out_size, void* d_ws,
                              size_t ws_size, hipStream_t stream) {
  (void)in_sizes;
  (void)n_in;
  (void)out_size;
  (void)ws_size;
  const float* x = (const float*)d_in[0];
  const float* qkv_w = (const float*)d_in[1];
  const float* o_w = (const float*)d_in[2];
  const float* o_scale = (const float*)d_in[3];
  const float* w1 = (const float*)d_in[4];
  const float* w2 = (const float*)d_in[5];
  const float* w2_scale = (const float*)d_in[6];
  float* out = (float*)d_out;

  const size_t TD = (size_t)T_SEQ * DMODEL;
  float* xn = (float*)d_ws;  // 16MB (reused as xn2)
  float* qkv = xn + TD;      // 48MB
  float* qh = qkv + 3 * TD;  // 16MB
  float* kh = qh + TD;       // 16MB
  float* vh = kh + TD;       // 16MB
  float* attn = vh + TD;     // 16MB
  float* x1 = attn + TD;     // 16MB
  float* hbuf = qkv;         // 64MB, reuses dead qkv+qh regions

  rmsnorm_kernel<<<T_SEQ, 256, 0, stream>>>(x, xn, DMODEL);
  gemm_wt_kernel<<<dim3(3 * DMODEL / BN, T_SEQ / BM), 256, 0, stream>>>(
      xn, qkv_w, nullptr, nullptr, qkv, T_SEQ, 3 * DMODEL, DMODEL, 0);
  qk_rope_kernel<<<dim3(T_SEQ, NH), 64, 0, stream>>>(qkv, qh, kh, vh);
  attn_kernel<<<dim3(NH, T_SEQ / 16), 32, 0, stream>>>(qh, kh, vh, attn);
  gemm_wt_kernel<<<dim3(DMODEL / BN, T_SEQ / BM), 256, 0, stream>>>(
      attn, o_w, o_scale, x, x1, T_SEQ, DMODEL, DMODEL, 0);
  rmsnorm_kernel<<<T_SEQ, 256, 0, stream>>>(x1, xn, DMODEL);
  gemm_wt_kernel<<<dim3(DMLP / BN, T_SEQ / BM), 256, 0, stream>>>(
      xn, w1, nullptr, nullptr, hbuf, T_SEQ, DMLP, DMODEL, 1);
  gemm_wt_kernel<<<dim3(DMODEL / BN, T_SEQ / BM), 256, 0, stream>>>(
      hbuf, w2, w2_scale, x1, out, T_SEQ, DMODEL, DMLP, 0);
}